// MotionMambaBlock_24816321036291
// MI455X (gfx1250) — compile-verified
//
#include <hip/hip_runtime.h>
#include <hip/hip_bf16.h>
#include <math.h>

// ---------------- problem constants ----------------
#define BB      2
#define LL      2048
#define DMODEL  512
#define DINNER  1024
#define DSTATE  16
#define DCONV   4
#define DTRANK  32
#define HIDDEN  2048
#define ML      (BB*LL)          // 4096 rows
#define LNEPS   1e-5f

typedef __attribute__((ext_vector_type(2))) float v2f;
typedef __attribute__((ext_vector_type(8))) float v8f;

// ---------------- device math helpers ----------------
__device__ __forceinline__ float silu_f(float x) {
    return x / (1.0f + expf(-x));
}
__device__ __forceinline__ float gelu_f(float x) {
    // exact gelu: 0.5*x*(1+erf(x/sqrt(2)))
    return 0.5f * x * (1.0f + erff(x * 0.70710678118654752440f));
}
__device__ __forceinline__ float softplus_f(float x) {
    return (x > 20.0f) ? x : log1pf(expf(x));
}

// ---------------- LayerNorm over 512 cols, one block per row ----------------
__global__ __launch_bounds__(128)
void layernorm_kernel(const float* __restrict__ x,
                      const float* __restrict__ w,
                      const float* __restrict__ b,
                      float* __restrict__ out)
{
    __shared__ float red[128];
    const int row = blockIdx.x;
    const int tid = threadIdx.x;
    const float* xr = x + (size_t)row * DMODEL;

    float v[4];
    float s = 0.0f;
    #pragma unroll
    for (int i = 0; i < 4; ++i) {
        v[i] = xr[tid + 128 * i];
        s += v[i];
    }
    red[tid] = s; __syncthreads();
    #pragma unroll
    for (int st = 64; st > 0; st >>= 1) {
        if (tid < st) red[tid] += red[tid + st];
        __syncthreads();
    }
    const float mean = red[0] * (1.0f / DMODEL);
    __syncthreads();

    float s2 = 0.0f;
    #pragma unroll
    for (int i = 0; i < 4; ++i) {
        float d = v[i] - mean;
        s2 += d * d;
    }
    red[tid] = s2; __syncthreads();
    #pragma unroll
    for (int st = 64; st > 0; st >>= 1) {
        if (tid < st) red[tid] += red[tid + st];
        __syncthreads();
    }
    const float rstd = rsqrtf(red[0] * (1.0f / DMODEL) + LNEPS);

    float* orow = out + (size_t)row * DMODEL;
    #pragma unroll
    for (int i = 0; i < 4; ++i) {
        int c = tid + 128 * i;
        orow[c] = (v[i] - mean) * rstd * w[c] + b[c];
    }
}

// =====================================================================
// Register-blocked fp32 WMMA GEMM: OUT = act(A[M,K] @ W[N,K]^T + bias)(+resid)
// Block = 128 threads = 4 waves arranged 2x2; block tile = 64(M) x 128(N).
// Each wave: 32x64 macro-tile = 2x4 grid of 16x16 WMMA tiles (8 accumulators).
// Per K-step(4): 2 A-frag + 4 B-frag loads feed 8 v_wmma -> 0.75 loads/wmma.
// grid.x = N/128, grid.y = M/64.  Requires K%4==0, N%128==0, M%64==0.
// ACT: 0=none, 1=gelu(exact), 2=softplus
// =====================================================================
template<int HAS_BIAS, int ACT, int HAS_RES>
__global__ __launch_bounds__(128)
void gemm_wmma_f32_blk(const float* __restrict__ A, int lda,
                       const float* __restrict__ W, int ldw,
                       const float* __restrict__ bias,
                       const float* __restrict__ resid, int ldr,
                       float* __restrict__ OUT, int ldo,
                       int K)
{
    const int wave = threadIdx.x >> 5;
    const int lane = threadIdx.x & 31;
    const int half = lane >> 4;      // selects K pair {0,1} vs {2,3} / M low/high
    const int lr   = lane & 15;

    const int wm = wave >> 1;        // 0..1
    const int wn = wave & 1;         // 0..1
    const int row0 = blockIdx.y * 64 + wm * 32;   // wave's first row
    const int col0 = blockIdx.x * 128 + wn * 64;  // wave's first col

    // fragment base pointers
    const float* ap[2];
    const float* wp[4];
    #pragma unroll
    for (int mi = 0; mi < 2; ++mi)
        ap[mi] = A + (size_t)(row0 + mi * 16 + lr) * lda + 2 * half;
    #pragma unroll
    for (int j = 0; j < 4; ++j)
        wp[j] = W + (size_t)(col0 + j * 16 + lr) * ldw + 2 * half;

    v8f acc[2][4];
    #pragma unroll
    for (int mi = 0; mi < 2; ++mi)
        #pragma unroll
        for (int j = 0; j < 4; ++j)
            acc[mi][j] = (v8f){0.f,0.f,0.f,0.f,0.f,0.f,0.f,0.f};

    for (int k0 = 0; k0 < K; k0 += 4) {
        v2f a[2], b[4];
        #pragma unroll
        for (int mi = 0; mi < 2; ++mi) a[mi] = *(const v2f*)(ap[mi] + k0);
        #pragma unroll
        for (int j = 0; j < 4; ++j)    b[j]  = *(const v2f*)(wp[j] + k0);
        #pragma unroll
        for (int mi = 0; mi < 2; ++mi)
            #pragma unroll
            for (int j = 0; j < 4; ++j)
                acc[mi][j] = __builtin_amdgcn_wmma_f32_16x16x4_f32(
                    false, a[mi], false, b[j], (short)0, acc[mi][j], false, false);
    }

    #pragma unroll
    for (int j = 0; j < 4; ++j) {
        const int col = col0 + j * 16 + lr;
        const float bv = HAS_BIAS ? bias[col] : 0.0f;
        #pragma unroll
        for (int mi = 0; mi < 2; ++mi) {
            #pragma unroll
            for (int r = 0; r < 8; ++r) {
                const int m = row0 + mi * 16 + r + 8 * half;
                float v = acc[mi][j][r] + bv;
                if (ACT == 1) v = gelu_f(v);
                if (ACT == 2) v = softplus_f(v);
                if (HAS_RES) v += resid[(size_t)m * ldr + col];
                OUT[(size_t)m * ldo + col] = v;
            }
        }
    }
}

// ---------------- simple (single-tile) WMMA GEMM, for N=64 (x_proj) --------
// grid.x = N/64, grid.y = M/16; one wave -> one 16x16 tile.
template<int HAS_BIAS, int ACT, int HAS_RES>
__global__ __launch_bounds__(128)
void gemm_wmma_f32(const float* __restrict__ A, int lda,
                   const float* __restrict__ W, int ldw,
                   const float* __restrict__ bias,
                   const float* __restrict__ resid, int ldr,
                   float* __restrict__ OUT, int ldo,
                   int K)
{
    const int wave = threadIdx.x >> 5;
    const int lane = threadIdx.x & 31;
    const int half = lane >> 4;
    const int lr   = lane & 15;

    const int row0 = blockIdx.y * 16;
    const int col  = (blockIdx.x * 4 + wave) * 16 + lr;

    const float* ap = A + (size_t)(row0 + lr) * lda + 2 * half;
    const float* wp = W + (size_t)col * ldw + 2 * half;

    v8f acc = {0.f,0.f,0.f,0.f,0.f,0.f,0.f,0.f};
    for (int k0 = 0; k0 < K; k0 += 4) {
        v2f a = *(const v2f*)(ap + k0);
        v2f b = *(const v2f*)(wp + k0);
        acc = __builtin_amdgcn_wmma_f32_16x16x4_f32(
            false, a, false, b, (short)0, acc, false, false);
    }

    const float bv = HAS_BIAS ? bias[col] : 0.0f;
    #pragma unroll
    for (int j = 0; j < 8; ++j) {
        const int m = row0 + j + 8 * half;
        float v = acc[j] + bv;
        if (ACT == 1) v = gelu_f(v);
        if (ACT == 2) v = softplus_f(v);
        if (HAS_RES) v += resid[(size_t)m * ldr + col];
        OUT[(size_t)m * ldo + col] = v;
    }
}

// ---------------- depthwise causal conv(k=4) + SiLU -------------------------
__global__ __launch_bounds__(256)
void conv_silu_kernel(const float* __restrict__ xz,
                      const float* __restrict__ cw,   // [1024,1,4]
                      const float* __restrict__ cb,   // [1024]
                      float* __restrict__ ut, int dir)
{
    const int t = blockIdx.x * 256 + threadIdx.x;   // total 2^22
    const int c = t & (DINNER - 1);
    const int l = (t >> 10) & (LL - 1);
    const int b = t >> 21;

    float acc = cb[c];
    #pragma unroll
    for (int j = 0; j < DCONV; ++j) {
        const int p = l - (DCONV - 1) + j;
        if (p >= 0) {
            const int lo = dir ? (LL - 1 - p) : p;
            acc += cw[c * DCONV + j] * xz[((size_t)(b * LL + lo)) * (2 * DINNER) + c];
        }
    }
    ut[((size_t)(b * LL + l)) * DINNER + c] = silu_f(acc);
}

// ---------------- selective scan --------------------------------------------
// grid: (DINNER/256, BB). Each thread owns one channel d with 16 states in regs.
// B/C tiles (cols 32..63 of x_dbl) staged into LDS 64 timesteps at a time.
__global__ __launch_bounds__(256)
void scan_kernel(const float* __restrict__ ut,     // [B,L,1024] (dir order)
                 const float* __restrict__ dt,     // [B,L,1024] (dir order)
                 const float* __restrict__ xdbl,   // [B,L,64]   (dir order)
                 const float* __restrict__ xz,     // [B,L,2048] original order
                 const float* __restrict__ A_log,  // [1024,16]
                 const float* __restrict__ Dp,     // [1024]
                 float* __restrict__ y, int dir)
{
    __shared__ float sB[64][DSTATE];
    __shared__ float sC[64][DSTATE];

    const int d = blockIdx.x * 256 + threadIdx.x;
    const int b = blockIdx.y;

    float A[DSTATE], h[DSTATE];
    #pragma unroll
    for (int n = 0; n < DSTATE; ++n) {
        A[n] = -expf(A_log[d * DSTATE + n]);
        h[n] = 0.0f;
    }
    const float Dd = Dp[d];

    for (int l0 = 0; l0 < LL; l0 += 64) {
        __syncthreads();
        for (int t = threadIdx.x; t < 64 * DSTATE; t += 256) {
            const int i = t >> 4, n = t & 15;
            const float* row = xdbl + ((size_t)(b * LL + l0 + i)) * (DTRANK + 2 * DSTATE);
            sB[i][n] = row[DTRANK + n];
            sC[i][n] = row[DTRANK + DSTATE + n];
        }
        __syncthreads();

        for (int i = 0; i < 64; ++i) {
            const int l = l0 + i;
            const size_t base = (size_t)(b * LL + l);
            const float dtv = dt[base * DINNER + d];
            const float uv  = ut[base * DINNER + d];
            const float du  = dtv * uv;
            float yv = 0.0f;
            #pragma unroll
            for (int n = 0; n < DSTATE; ++n) {
                const float dA = expf(dtv * A[n]);
                h[n] = dA * h[n] + du * sB[i][n];
                yv += h[n] * sC[i][n];
            }
            const int lz = dir ? (LL - 1 - l) : l;
            const float zv = xz[((size_t)(b * LL + lz)) * (2 * DINNER) + DINNER + d];
            y[base * DINNER + d] = (yv + Dd * uv) * silu_f(zv);
        }
    }
}

// ---------------- combine forward + time-flipped backward -------------------
__global__ __launch_bounds__(256)
void combine_kernel(const float* __restrict__ yf,
                    const float* __restrict__ yb,
                    float* __restrict__ out)
{
    const int t = blockIdx.x * 256 + threadIdx.x;
    const int d = t & (DINNER - 1);
    const int l = (t >> 10) & (LL - 1);
    const int b = t >> 21;
    const float vf = yf[((size_t)(b * LL + l)) * DINNER + d];
    const float vb = yb[((size_t)(b * LL + (LL - 1 - l))) * DINNER + d];
    out[((size_t)(b * LL + l)) * DINNER + d] = 0.5f * (vf + vb);
}

// ---------------- host launcher ---------------------------------------------
extern "C" void kernel_launch(void* const* d_in, const int* in_sizes, int n_in,
                              void* d_out, int out_size, void* d_ws, size_t ws_size,
                              hipStream_t stream) {
    (void)in_sizes; (void)n_in; (void)out_size; (void)ws_size;

    const float* x          = (const float*)d_in[0];
    const float* ln1_w      = (const float*)d_in[1];
    const float* ln1_b      = (const float*)d_in[2];
    const float* in_proj_w  = (const float*)d_in[3];
    const float* conv_w     = (const float*)d_in[4];
    const float* conv_b     = (const float*)d_in[5];
    const float* x_proj_w   = (const float*)d_in[6];
    const float* dt_proj_w  = (const float*)d_in[7];
    const float* dt_proj_b  = (const float*)d_in[8];
    const float* A_log      = (const float*)d_in[9];
    const float* D_param    = (const float*)d_in[10];
    const float* conv_w_b   = (const float*)d_in[11];
    const float* conv_b_b   = (const float*)d_in[12];
    const float* x_proj_w_b = (const float*)d_in[13];
    const float* dt_proj_w_b= (const float*)d_in[14];
    const float* dt_proj_b_b= (const float*)d_in[15];
    const float* A_log_b    = (const float*)d_in[16];
    const float* D_param_b  = (const float*)d_in[17];
    const float* out_proj_w = (const float*)d_in[18];
    const float* ln2_w      = (const float*)d_in[19];
    const float* ln2_b      = (const float*)d_in[20];
    const float* fc1_w      = (const float*)d_in[21];
    const float* fc1_b      = (const float*)d_in[22];
    const float* fc2_w      = (const float*)d_in[23];
    const float* fc2_b      = (const float*)d_in[24];
    float* out = (float*)d_out;

    // ---- workspace layout (floats); high-water ~34.5M floats (~138 MB) ----
    float* ws     = (float*)d_ws;
    float* xn     = ws;                                 // 4096*512
    float* xz     = xn     + (size_t)ML * DMODEL;       // 4096*2048
    float* ut_f   = xz     + (size_t)ML * 2 * DINNER;   // 4096*1024
    float* ut_b   = ut_f   + (size_t)ML * DINNER;
    float* xdbl_f = ut_b   + (size_t)ML * DINNER;       // 4096*64
    float* xdbl_b = xdbl_f + (size_t)ML * 64;
    float* dt_f   = xdbl_b + (size_t)ML * 64;           // 4096*1024
    float* dt_b   = dt_f   + (size_t)ML * DINNER;
    float* y_f    = dt_b   + (size_t)ML * DINNER;
    float* y_b    = y_f    + (size_t)ML * DINNER;
    // reuse (lifetimes disjoint):
    float* ycomb  = ut_f;   // ut_f dead after scans
    float* x1     = xn;     // xn dead after in_proj
    float* x2     = dt_f;   // dt_f dead after scan_f/scan_b
    float* hbuf   = xz;     // xz dead after scans (gating)

    const int elems = BB * LL * DINNER;                 // 2^22

    // 1. LN1
    layernorm_kernel<<<ML, 128, 0, stream>>>(x, ln1_w, ln1_b, xn);

    // 2. in_proj: xz[M,2048] = xn @ in_proj_w^T
    gemm_wmma_f32_blk<0,0,0><<<dim3((2*DINNER)/128, ML/64), 128, 0, stream>>>(
        xn, DMODEL, in_proj_w, DMODEL, nullptr, nullptr, 0, xz, 2*DINNER, DMODEL);

    // 3. depthwise conv + SiLU, both directions
    conv_silu_kernel<<<elems/256, 256, 0, stream>>>(xz, conv_w,   conv_b,   ut_f, 0);
    conv_silu_kernel<<<elems/256, 256, 0, stream>>>(xz, conv_w_b, conv_b_b, ut_b, 1);

    // 4. x_proj: x_dbl[M,64] = ut @ x_proj_w^T   (N=64 -> single-tile kernel)
    gemm_wmma_f32<0,0,0><<<dim3(64/64, ML/16), 128, 0, stream>>>(
        ut_f, DINNER, x_proj_w,   DINNER, nullptr, nullptr, 0, xdbl_f, 64, DINNER);
    gemm_wmma_f32<0,0,0><<<dim3(64/64, ML/16), 128, 0, stream>>>(
        ut_b, DINNER, x_proj_w_b, DINNER, nullptr, nullptr, 0, xdbl_b, 64, DINNER);

    // 5. dt = softplus(x_dbl[:, :32] @ dt_proj_w^T + dt_bias)
    gemm_wmma_f32_blk<1,2,0><<<dim3(DINNER/128, ML/64), 128, 0, stream>>>(
        xdbl_f, 64, dt_proj_w,   DTRANK, dt_proj_b,   nullptr, 0, dt_f, DINNER, DTRANK);
    gemm_wmma_f32_blk<1,2,0><<<dim3(DINNER/128, ML/64), 128, 0, stream>>>(
        xdbl_b, 64, dt_proj_w_b, DTRANK, dt_proj_b_b, nullptr, 0, dt_b, DINNER, DTRANK);

    // 6. selective scans (serial in L; channel-parallel) + D*u skip + SiLU(z) gate
    scan_kernel<<<dim3(DINNER/256, BB), 256, 0, stream>>>(
        ut_f, dt_f, xdbl_f, xz, A_log,   D_param,   y_f, 0);
    scan_kernel<<<dim3(DINNER/256, BB), 256, 0, stream>>>(
        ut_b, dt_b, xdbl_b, xz, A_log_b, D_param_b, y_b, 1);

    // 7. y = 0.5*(y_f + flip(y_b))
    combine_kernel<<<elems/256, 256, 0, stream>>>(y_f, y_b, ycomb);

    // 8. x1 = x + y @ out_proj_w^T
    gemm_wmma_f32_blk<0,0,1><<<dim3(DMODEL/128, ML/64), 128, 0, stream>>>(
        ycomb, DINNER, out_proj_w, DINNER, nullptr, x, DMODEL, x1, DMODEL, DINNER);

    // 9. LN2
    layernorm_kernel<<<ML, 128, 0, stream>>>(x1, ln2_w, ln2_b, x2);

    // 10. h = gelu(x2 @ fc1_w^T + fc1_b)
    gemm_wmma_f32_blk<1,1,0><<<dim3(HIDDEN/128, ML/64), 128, 0, stream>>>(
        x2, DMODEL, fc1_w, DMODEL, fc1_b, nullptr, 0, hbuf, HIDDEN, DMODEL);

    // 11. out = x1 + h @ fc2_w^T + fc2_b
    gemm_wmma_f32_blk<1,0,1><<<dim3(DMODEL/128, ML/64), 128, 0, stream>>>(
        hbuf, HIDDEN, fc2_w, HIDDEN, fc2_b, x1, DMODEL, out, DMODEL, HIDDEN);
}